// Features2FeaturesResidual_79328045957691
// MI455X (gfx1250) — compile-verified
//
#include <hip/hip_runtime.h>

// Problem constants (match reference).
#define NNODES 50000
#define NEDGES 600000
#define FDIM   128

typedef float v2f __attribute__((ext_vector_type(2)));
typedef float v8f __attribute__((ext_vector_type(8)));

// ---------------------------------------------------------------------------
// Degree kernels
// ---------------------------------------------------------------------------
__global__ void zero_f32(float* __restrict__ p, int n) {
    int i = blockIdx.x * blockDim.x + threadIdx.x;
    if (i < n) p[i] = 0.0f;
}

__global__ void deg_accum(const int* __restrict__ edges, float* __restrict__ deg, int ne) {
    int e = blockIdx.x * blockDim.x + threadIdx.x;
    if (e < ne) {
        int s = edges[2 * e + 0];
        int d = edges[2 * e + 1];
        atomicAdd(deg + d, 1.0f);
        atomicAdd(deg + s, 1.0f);
    }
}

__global__ void deg_recip(float* __restrict__ deg, int n) {
    int i = blockIdx.x * blockDim.x + threadIdx.x;
    if (i < n) deg[i] = 1.0f / fmaxf(deg[i], 1.0f);
}

// ---------------------------------------------------------------------------
// WMMA GEMM: Out[m,n] = sum_k X[m,k] * W[n,k]  (+ bias[n] if bias != nullptr)
// X: nrows x 128 row-major, W: 128 x 128 row-major (we multiply by W^T).
// Block = 256 threads = 8 waves. Block computes a 16-row x 128-col stripe;
// wave w owns the 16-col tile [16w, 16w+16). fp32 path: v_wmma_f32_16x16x4_f32.
//
// Fragment layouts (CDNA5 ISA 7.12.2, wave32):
//   A 16x4 : lanes 0-15 -> M = lane, v0=K0 v1=K1 ; lanes 16-31 -> v0=K2 v1=K3
//   B 4x16 : lanes 0-15 -> N = lane, v0=K0 v1=K1 ; lanes 16-31 -> v0=K2 v1=K3
//   C 16x16: VGPR r -> M = r (lanes 0-15), M = r+8 (lanes 16-31), N = lane&15
// ---------------------------------------------------------------------------
__global__ __launch_bounds__(256) void gcn_gemm_xwT(
    const float* __restrict__ X,
    const float* __restrict__ W,
    const float* __restrict__ bias,
    float* __restrict__ Out)
{
    __shared__ __align__(16) float sA[16 * 132];   // 16x128 X tile, padded stride

    const int tid  = threadIdx.x;
    const int wave = tid >> 5;          // 0..7  -> output column tile
    const int lane = tid & 31;
    const int row0 = blockIdx.x * 16;

    // Cooperative stage of the 16x128 X tile into LDS (8 floats/thread).
    {
        const int r = tid >> 4;          // 0..15
        const int c = (tid & 15) * 8;    // 0..120
        const float4* src = (const float4*)(X + (size_t)(row0 + r) * FDIM + c);
        float4 a0 = src[0];
        float4 a1 = src[1];
        float* dst = sA + r * 132 + c;
        *(float4*)(dst + 0) = a0;
        *(float4*)(dst + 4) = a1;
    }
    __syncthreads();

    const int hi  = lane >> 4;           // 0: K0/K1 half, 1: K2/K3 half
    const int lo  = lane & 15;
    const int col = wave * 16 + lo;      // this lane's output column
    const float* Wrow = W + (size_t)col * FDIM;   // row `col` of W == column of W^T

    v8f acc = {};
#pragma unroll 8
    for (int k = 0; k < FDIM; k += 4) {
        v2f a = *(const v2f*)(sA + lo * 132 + k + 2 * hi);    // ds_load_b64
        v2f b = *(const v2f*)(Wrow + k + 2 * hi);             // global_load_b64
        acc = __builtin_amdgcn_wmma_f32_16x16x4_f32(
            /*neg_a=*/false, a, /*neg_b=*/false, b,
            /*c_mod=*/(short)0, acc, /*reuse_a=*/false, /*reuse_b=*/false);
    }

    const float bv = bias ? bias[col] : 0.0f;
    float* outBase = Out + (size_t)(row0 + 8 * hi) * FDIM + col;
#pragma unroll
    for (int r = 0; r < 8; ++r)
        outBase[(size_t)r * FDIM] = acc[r] + bv;
}

// ---------------------------------------------------------------------------
// Edge scatter: one wave per edge, lane owns 4 feature channels.
// o[dst] += h[src] * invdeg[dst] ; o[src] += h[dst] * invdeg[src]
// (degree normalization folded into the atomic contribution, so `o` can be
//  pre-initialized with x@W0^T + b by the GEMM above.)
// ---------------------------------------------------------------------------
__global__ __launch_bounds__(256) void gcn_scatter(
    const int* __restrict__ edges,
    const float* __restrict__ h,
    const float* __restrict__ invdeg,
    float* __restrict__ o,
    int ne)
{
    const int gwave = (blockIdx.x * blockDim.x + threadIdx.x) >> 5;
    const int lane  = threadIdx.x & 31;
    if (gwave >= ne) return;

    const int s = edges[2 * gwave + 0];
    const int d = edges[2 * gwave + 1];
    const float ids = invdeg[s];
    const float idd = invdeg[d];
    const int c = lane * 4;

    const float4 hs = *(const float4*)(h + (size_t)s * FDIM + c);
    const float4 hd = *(const float4*)(h + (size_t)d * FDIM + c);

    float* od = o + (size_t)d * FDIM + c;
    float* os = o + (size_t)s * FDIM + c;
    atomicAdd(od + 0, hs.x * idd);
    atomicAdd(od + 1, hs.y * idd);
    atomicAdd(od + 2, hs.z * idd);
    atomicAdd(od + 3, hs.w * idd);
    atomicAdd(os + 0, hd.x * ids);
    atomicAdd(os + 1, hd.y * ids);
    atomicAdd(os + 2, hd.z * ids);
    atomicAdd(os + 3, hd.w * ids);
}

// ---------------------------------------------------------------------------
// Elementwise epilogues
// ---------------------------------------------------------------------------
__global__ void relu_inplace(float* __restrict__ p, int n4) {
    int i = blockIdx.x * blockDim.x + threadIdx.x;
    if (i < n4) {
        float4 v = ((float4*)p)[i];
        v.x = fmaxf(v.x, 0.0f);
        v.y = fmaxf(v.y, 0.0f);
        v.z = fmaxf(v.z, 0.0f);
        v.w = fmaxf(v.w, 0.0f);
        ((float4*)p)[i] = v;
    }
}

__global__ void residual_relu_out(const float* __restrict__ o,
                                  const float* __restrict__ res,
                                  float* __restrict__ out, int n4) {
    int i = blockIdx.x * blockDim.x + threadIdx.x;
    if (i < n4) {
        float4 a = ((const float4*)o)[i];
        float4 b = ((const float4*)res)[i];
        float4 v;
        v.x = fmaxf(a.x + b.x, 0.0f);
        v.y = fmaxf(a.y + b.y, 0.0f);
        v.z = fmaxf(a.z + b.z, 0.0f);
        v.w = fmaxf(a.w + b.w, 0.0f);
        ((float4*)out)[i] = v;
    }
}

// ---------------------------------------------------------------------------
// Host orchestration
// ---------------------------------------------------------------------------
static void run_layer(const float* X, const float* W1, const float* W0,
                      const float* b, float* Hbuf, float* Obuf,
                      const int* edges, const float* invdeg, hipStream_t stream)
{
    const int gemmBlocks = NNODES / 16;                 // 3125 (exact)
    gcn_gemm_xwT<<<gemmBlocks, 256, 0, stream>>>(X, W1, nullptr, Hbuf);  // h = x@W1^T
    gcn_gemm_xwT<<<gemmBlocks, 256, 0, stream>>>(X, W0, b, Obuf);        // o = x@W0^T + b
    const int scatBlocks = (NEDGES + 7) / 8;            // 8 waves/block
    gcn_scatter<<<scatBlocks, 256, 0, stream>>>(edges, Hbuf, invdeg, Obuf, NEDGES);
}

extern "C" void kernel_launch(void* const* d_in, const int* in_sizes, int n_in,
                              void* d_out, int out_size, void* d_ws, size_t ws_size,
                              hipStream_t stream)
{
    const float* features = (const float*)d_in[0];   // N x D
    const int*   edges    = (const int*)d_in[1];     // E x 2
    // d_in[2] = dis (unused by reference)
    const float* W0_first = (const float*)d_in[3];
    const float* W1_first = (const float*)d_in[4];
    const float* b_first  = (const float*)d_in[5];
    const float* W0_h     = (const float*)d_in[6];   // H x D x D
    const float* W1_h     = (const float*)d_in[7];
    const float* b_h      = (const float*)d_in[8];   // H x D

    float* ws   = (float*)d_ws;
    float* deg  = ws;                                // N floats (-> 1/deg)
    float* buf0 = ws + 50176;                        // N*D each, rotated
    float* buf1 = buf0 + (size_t)NNODES * FDIM;
    float* buf2 = buf1 + (size_t)NNODES * FDIM;

    // Degrees (re-zeroed every call: deterministic, no persistent ws state).
    zero_f32<<<(NNODES + 255) / 256, 256, 0, stream>>>(deg, NNODES);
    deg_accum<<<(NEDGES + 255) / 256, 256, 0, stream>>>(edges, deg, NEDGES);
    deg_recip<<<(NNODES + 255) / 256, 256, 0, stream>>>(deg, NNODES);

    const int n4 = NNODES * FDIM / 4;
    const size_t DD = (size_t)FDIM * FDIM;

    // Layer 0: f = gconv(features)              (no activation)
    run_layer(features, W1_first, W0_first, b_first, buf0, buf1, edges, deg, stream);
    // Layer 1: f = relu(gconv(f))
    run_layer(buf1, W1_h + 0 * DD, W0_h + 0 * DD, b_h + 0 * FDIM, buf0, buf2, edges, deg, stream);
    relu_inplace<<<(n4 + 255) / 256, 256, 0, stream>>>(buf2, n4);
    // Layer 2: f = relu(gconv(f))
    run_layer(buf2, W1_h + 1 * DD, W0_h + 1 * DD, b_h + 1 * FDIM, buf0, buf1, edges, deg, stream);
    relu_inplace<<<(n4 + 255) / 256, 256, 0, stream>>>(buf1, n4);
    // Layer 3: out = relu(gconv(f) + features)
    run_layer(buf1, W1_h + 2 * DD, W0_h + 2 * DD, b_h + 2 * FDIM, buf0, buf2, edges, deg, stream);
    residual_relu_out<<<(n4 + 255) / 256, 256, 0, stream>>>(buf2, features, (float*)d_out, n4);
}